// GlobalUpdate_53730040873193
// MI455X (gfx1250) — compile-verified
//
#include <hip/hip_runtime.h>
#include <hip/hip_bf16.h>

typedef __attribute__((ext_vector_type(2))) float v2f;
typedef __attribute__((ext_vector_type(8))) float v8f;

#define DM   128          // node/edge/global feature dim
#define HM   512          // MLP hidden (4*D)
#define K3M  384          // 3*D concat width

static __device__ __forceinline__ v8f wmma_f32_16x16x4(v2f a, v2f b, v8f c) {
    // D = A(16x4) * B(4x16) + C(16x16), all f32, wave32
    return __builtin_amdgcn_wmma_f32_16x16x4_f32(
        /*neg_a=*/false, a, /*neg_b=*/false, b,
        /*c_mod=*/(short)0, c, /*reuse_a=*/false, /*reuse_b=*/false);
}

// ======================= fast path: binned, atomic-free reductions ==============

__global__ void gu_zero_i32_kernel(int* p, int n) {
    int i = blockIdx.x * blockDim.x + threadIdx.x;
    int stride = gridDim.x * blockDim.x;
    for (; i < n; i += stride) p[i] = 0;
}

// histogram of edge graph-ids + cache gid per edge
__global__ void gu_hist_kernel(const int* __restrict__ edge_src,
                               const int* __restrict__ batch,
                               int* __restrict__ edge_gid,
                               int* __restrict__ bin_count, int nE) {
    int i = blockIdx.x * blockDim.x + threadIdx.x;
    int stride = gridDim.x * blockDim.x;
    for (int e = i; e < nE; e += stride) {
        int g = batch[edge_src[e]];
        edge_gid[e] = g;
        atomicAdd(bin_count + g, 1);
    }
}

// exclusive scan over <=1024 bins (single block, Hillis-Steele in LDS)
__global__ __launch_bounds__(1024) void gu_scan_kernel(const int* __restrict__ bin_count,
                                                       int* __restrict__ bin_start,
                                                       int* __restrict__ cursor, int G) {
    __shared__ int tmp[1024];
    int t = threadIdx.x;
    int own = (t < G) ? bin_count[t] : 0;
    tmp[t] = own;
    __syncthreads();
    for (int off = 1; off < 1024; off <<= 1) {
        int v = (t >= off) ? tmp[t - off] : 0;
        __syncthreads();
        tmp[t] += v;
        __syncthreads();
    }
    if (t < G) {
        int excl = tmp[t] - own;
        bin_start[t] = excl;
        cursor[t]    = excl;
    }
}

// scatter edge ids into graph-binned order
__global__ void gu_binplace_kernel(const int* __restrict__ edge_gid,
                                   int* __restrict__ cursor,
                                   int* __restrict__ edge_order, int nE) {
    int i = blockIdx.x * blockDim.x + threadIdx.x;
    int stride = gridDim.x * blockDim.x;
    for (int e = i; e < nE; e += stride) {
        int g = edge_gid[e];
        int pos = atomicAdd(cursor + g, 1);
        edge_order[pos] = e;
    }
}

// one workgroup per graph: dense reduction of its binned edge rows -> edge_mean
__global__ __launch_bounds__(256) void gu_edge_reduce_kernel(
    const float* __restrict__ edge_attr,
    const int* __restrict__ edge_order,
    const int* __restrict__ bin_start,
    const int* __restrict__ bin_count,
    float* __restrict__ edge_mean) {
    __shared__ float part[8][128];
    const int g    = blockIdx.x;
    const int wave = threadIdx.x >> 5;
    const int lane = threadIdx.x & 31;
    const int lo   = bin_start[g];
    const int cnt  = bin_count[g];
    float ax = 0.f, ay = 0.f, az = 0.f, aw = 0.f;
    for (int i = lo + wave; i < lo + cnt; i += 8) {
        int e = edge_order[i];
        const float4 v = *(const float4*)(edge_attr + (size_t)e * DM + lane * 4);
        ax += v.x; ay += v.y; az += v.z; aw += v.w;
    }
    part[wave][lane * 4 + 0] = ax;
    part[wave][lane * 4 + 1] = ay;
    part[wave][lane * 4 + 2] = az;
    part[wave][lane * 4 + 3] = aw;
    __syncthreads();
    if (threadIdx.x < DM) {
        float s = 0.f;
        #pragma unroll
        for (int w = 0; w < 8; ++w) s += part[w][threadIdx.x];
        edge_mean[(size_t)g * DM + threadIdx.x] = s / fmaxf((float)cnt, 1.0f);
    }
}

static __device__ __forceinline__ int lower_bound_i32(const int* a, int n, int key) {
    int lo = 0, hi = n;
    while (lo < hi) {
        int mid = (lo + hi) >> 1;
        if (a[mid] < key) lo = mid + 1; else hi = mid;
    }
    return lo;
}

// one workgroup per graph: batch is sorted -> binary-search range, stream rows
__global__ __launch_bounds__(256) void gu_node_reduce_kernel(
    const float* __restrict__ x,
    const int* __restrict__ batch,
    float* __restrict__ node_mean, int nN) {
    __shared__ float part[8][128];
    const int g    = blockIdx.x;
    const int wave = threadIdx.x >> 5;
    const int lane = threadIdx.x & 31;
    const int lo   = lower_bound_i32(batch, nN, g);
    const int hi   = lower_bound_i32(batch, nN, g + 1);
    float ax = 0.f, ay = 0.f, az = 0.f, aw = 0.f;
    for (int n = lo + wave; n < hi; n += 8) {
        const float4 v = *(const float4*)(x + (size_t)n * DM + lane * 4);
        ax += v.x; ay += v.y; az += v.z; aw += v.w;
    }
    part[wave][lane * 4 + 0] = ax;
    part[wave][lane * 4 + 1] = ay;
    part[wave][lane * 4 + 2] = az;
    part[wave][lane * 4 + 3] = aw;
    __syncthreads();
    if (threadIdx.x < DM) {
        float s = 0.f;
        #pragma unroll
        for (int w = 0; w < 8; ++w) s += part[w][threadIdx.x];
        node_mean[(size_t)g * DM + threadIdx.x] = s / fmaxf((float)(hi - lo), 1.0f);
    }
}

// ======================= fallback path: global fp32 atomics =====================

__global__ void gu_zero_f32_kernel(float* ws, int n) {
    int i = blockIdx.x * blockDim.x + threadIdx.x;
    int stride = gridDim.x * blockDim.x;
    for (; i < n; i += stride) ws[i] = 0.0f;
}

__global__ void gu_edge_scatter_kernel(const float* __restrict__ edge_attr,
                                       const int* __restrict__ edge_src,
                                       const int* __restrict__ batch,
                                       float* __restrict__ edge_sum,
                                       float* __restrict__ edge_cnt, int nE) {
    int gtid   = blockIdx.x * blockDim.x + threadIdx.x;
    int wave   = gtid >> 5;
    int lane   = threadIdx.x & 31;
    int nWaves = (gridDim.x * blockDim.x) >> 5;
    for (int e = wave; e < nE; e += nWaves) {
        int g = batch[edge_src[e]];
        const float4 v = *(const float4*)(edge_attr + (size_t)e * DM + lane * 4);
        float* dst = edge_sum + (size_t)g * DM + lane * 4;
        atomicAdd(dst + 0, v.x);
        atomicAdd(dst + 1, v.y);
        atomicAdd(dst + 2, v.z);
        atomicAdd(dst + 3, v.w);
        if (lane == 0) atomicAdd(edge_cnt + g, 1.0f);
    }
}

__global__ void gu_node_scatter_kernel(const float* __restrict__ x,
                                       const int* __restrict__ batch,
                                       float* __restrict__ node_sum,
                                       float* __restrict__ node_cnt, int nN) {
    int gtid   = blockIdx.x * blockDim.x + threadIdx.x;
    int wave   = gtid >> 5;
    int lane   = threadIdx.x & 31;
    int nWaves = (gridDim.x * blockDim.x) >> 5;
    for (int n = wave; n < nN; n += nWaves) {
        int g = batch[n];
        const float4 v = *(const float4*)(x + (size_t)n * DM + lane * 4);
        float* dst = node_sum + (size_t)g * DM + lane * 4;
        atomicAdd(dst + 0, v.x);
        atomicAdd(dst + 1, v.y);
        atomicAdd(dst + 2, v.z);
        atomicAdd(dst + 3, v.w);
        if (lane == 0) atomicAdd(node_cnt + g, 1.0f);
    }
}

// =============== fused MLP (WMMA f32) + residual + LayerNorm ====================
// one block (8 waves / 256 threads) handles 16 graph rows; cnt ptrs nullable
__global__ __launch_bounds__(256) void gu_mlp_ln_kernel(
    const float* __restrict__ u,
    const float* __restrict__ W1, const float* __restrict__ b1,
    const float* __restrict__ W2, const float* __restrict__ b2,
    const float* __restrict__ gamma, const float* __restrict__ beta,
    const float* __restrict__ edge_acc, const float* __restrict__ node_acc,
    const float* edge_cnt, const float* node_cnt,
    float* __restrict__ y) {

    // LDS: inT[16][385] aliased later by outT[16][129]; hT[16][513]
    __shared__ float smem[16 * 385 + 16 * 513];
    __shared__ float mu_s[16], rs_s[16];
    float (*inT)[385]  = (float(*)[385])smem;
    float (*hT)[513]   = (float(*)[513])(smem + 16 * 385);
    float (*outT)[129] = (float(*)[129])smem;   // reuses inT space after GEMM1

    const int row0 = blockIdx.x * 16;
    const int tid  = threadIdx.x;
    const int wave = tid >> 5;
    const int lane = tid & 31;
    const int half = lane >> 4;
    const int mrow = lane & 15;

    // ---- stage concat features [u | edge_mean | node_mean] -> inT[16][384]
    for (int i = tid; i < 16 * K3M; i += 256) {
        int r = i / K3M, c = i % K3M;
        int g = row0 + r;
        float v;
        if (c < DM) {
            v = u[(size_t)g * DM + c];
        } else if (c < 2 * DM) {
            v = edge_acc[(size_t)g * DM + (c - DM)];
            if (edge_cnt) v /= fmaxf(edge_cnt[g], 1.0f);
        } else {
            v = node_acc[(size_t)g * DM + (c - 2 * DM)];
            if (node_cnt) v /= fmaxf(node_cnt[g], 1.0f);
        }
        inT[r][c] = v;
    }
    __syncthreads();

    // ---- GEMM1: h[16,512] = relu(inT @ W1 + b1); 4 N-tiles per wave
    for (int t = 0; t < 4; ++t) {
        int nbase = (wave * 4 + t) * 16;
        v8f acc = {};
        for (int k = 0; k < K3M; k += 4) {
            v2f a, b;
            a.x = inT[mrow][k + 2 * half + 0];   // A 16x4: lanes16-31 hold K=2,3
            a.y = inT[mrow][k + 2 * half + 1];
            b.x = W1[(size_t)(k + 2 * half + 0) * HM + nbase + mrow];
            b.y = W1[(size_t)(k + 2 * half + 1) * HM + nbase + mrow];
            acc = wmma_f32_16x16x4(a, b, acc);
        }
        int col = nbase + mrow;
        float bias = b1[col];
        #pragma unroll
        for (int r = 0; r < 8; ++r) {
            int rr = r + 8 * half;               // C/D: VGPR r -> M=r / M=r+8
            hT[rr][col] = fmaxf(acc[r] + bias, 0.0f);
        }
    }
    __syncthreads();

    // ---- GEMM2: out[16,128] = hT @ W2 + b2 + u ; 1 N-tile per wave
    {
        int nbase = wave * 16;
        v8f acc = {};
        for (int k = 0; k < HM; k += 4) {
            v2f a, b;
            a.x = hT[mrow][k + 2 * half + 0];
            a.y = hT[mrow][k + 2 * half + 1];
            b.x = W2[(size_t)(k + 2 * half + 0) * DM + nbase + mrow];
            b.y = W2[(size_t)(k + 2 * half + 1) * DM + nbase + mrow];
            acc = wmma_f32_16x16x4(a, b, acc);
        }
        int col = nbase + mrow;
        float bias = b2[col];
        #pragma unroll
        for (int r = 0; r < 8; ++r) {
            int rr = r + 8 * half;
            outT[rr][col] = acc[r] + bias + u[(size_t)(row0 + rr) * DM + col];
        }
    }
    __syncthreads();

    // ---- LayerNorm (rows are tiny: 128 cols)
    if (tid < 16) {
        float s = 0.0f;
        for (int c = 0; c < DM; ++c) s += outT[tid][c];
        float mu = s * (1.0f / DM);
        float var = 0.0f;
        for (int c = 0; c < DM; ++c) { float d = outT[tid][c] - mu; var += d * d; }
        mu_s[tid] = mu;
        rs_s[tid] = rsqrtf(var * (1.0f / DM) + 1e-5f);
    }
    __syncthreads();

    for (int i = tid; i < 16 * DM; i += 256) {
        int r = i / DM, c = i % DM;
        y[(size_t)(row0 + r) * DM + c] =
            (outT[r][c] - mu_s[r]) * rs_s[r] * gamma[c] + beta[c];
    }
}

// ================================ host launcher =================================
extern "C" void kernel_launch(void* const* d_in, const int* in_sizes, int n_in,
                              void* d_out, int out_size, void* d_ws, size_t ws_size,
                              hipStream_t stream) {
    const float* x         = (const float*)d_in[0];
    const float* edge_attr = (const float*)d_in[1];
    const float* u         = (const float*)d_in[2];
    const float* W1        = (const float*)d_in[3];
    const float* b1        = (const float*)d_in[4];
    const float* W2        = (const float*)d_in[5];
    const float* b2        = (const float*)d_in[6];
    const float* gamma     = (const float*)d_in[7];
    const float* beta      = (const float*)d_in[8];
    const int*   edge_idx  = (const int*)d_in[9];   // [2, nE]; row 0 = sources
    const int*   batch     = (const int*)d_in[10];  // [nN], sorted

    const int G  = in_sizes[2] / DM;
    const int nN = in_sizes[0] / DM;
    const int nE = in_sizes[1] / DM;
    const int* edge_src = edge_idx;

    (void)n_in; (void)out_size;

    const size_t fast_need = (size_t)(2 * G * DM) * 4         // edge_mean + node_mean
                           + (size_t)(3 * G + 1) * 4          // bin_count, bin_start, cursor
                           + (size_t)(2 * nE) * 4;            // edge_gid, edge_order

    if (G <= 1024 && ws_size >= fast_need) {
        // -------- fast path: atomic-free streaming reductions --------
        float* edge_mean = (float*)d_ws;                 // G*DM
        float* node_mean = edge_mean + (size_t)G * DM;   // G*DM
        int*   bin_count = (int*)(node_mean + (size_t)G * DM);  // G
        int*   bin_start = bin_count + G;                // G (+1 pad)
        int*   cursor    = bin_start + (G + 1);          // G
        int*   edge_gid  = cursor + G;                   // nE
        int*   edge_order= edge_gid + nE;                // nE

        gu_zero_i32_kernel<<<8, 256, 0, stream>>>(bin_count, 3 * G + 1);
        gu_hist_kernel<<<1024, 256, 0, stream>>>(edge_src, batch, edge_gid, bin_count, nE);
        gu_scan_kernel<<<1, 1024, 0, stream>>>(bin_count, bin_start, cursor, G);
        gu_binplace_kernel<<<1024, 256, 0, stream>>>(edge_gid, cursor, edge_order, nE);
        gu_edge_reduce_kernel<<<G, 256, 0, stream>>>(edge_attr, edge_order,
                                                     bin_start, bin_count, edge_mean);
        gu_node_reduce_kernel<<<G, 256, 0, stream>>>(x, batch, node_mean, nN);
        gu_mlp_ln_kernel<<<G / 16, 256, 0, stream>>>(u, W1, b1, W2, b2, gamma, beta,
                                                     edge_mean, node_mean,
                                                     nullptr, nullptr, (float*)d_out);
    } else {
        // -------- fallback: fp32 global atomics --------
        float* ws       = (float*)d_ws;
        float* edge_sum = ws;
        float* node_sum = ws + (size_t)G * DM;
        float* edge_cnt = ws + (size_t)2 * G * DM;
        float* node_cnt = edge_cnt + G;
        const int nAccum = 2 * G * DM + 2 * G;

        gu_zero_f32_kernel<<<256, 256, 0, stream>>>(ws, nAccum);
        gu_edge_scatter_kernel<<<1024, 256, 0, stream>>>(edge_attr, edge_src, batch,
                                                         edge_sum, edge_cnt, nE);
        gu_node_scatter_kernel<<<1024, 256, 0, stream>>>(x, batch, node_sum, node_cnt, nN);
        gu_mlp_ln_kernel<<<G / 16, 256, 0, stream>>>(u, W1, b1, W2, b2, gamma, beta,
                                                     edge_sum, node_sum,
                                                     edge_cnt, node_cnt, (float*)d_out);
    }
}